// RV2BEV_61469571940657
// MI455X (gfx1250) — compile-verified
//
#include <hip/hip_runtime.h>
#include <hip/hip_bf16.h>
#include <math.h>

// ---------------------------------------------------------------------------
// RV2BEV resample:  out[b,c,yy,xx] = bilerp of rv_feat[b,c, 31..32, x0..x0+1]
//   - py == 31.5 exactly (gy == 0) => wy == 0.5; only rows 31,32 are read.
//   - px depends only on (yy,xx): one 262144-entry fp32 table shared by all
//     256 (b,c) images; x0 = floor(px), wx = px - x0 (reference _PX is fp32).
//   - ref_bev input is unused by the reference output.
// Pipeline (all per-launch, deterministic):
//   K1 coords : fp64 px table -> d_ws[0 .. 1MB)
//   K2 blend  : 0.5*(row31+row32) per image -> d_ws[1MB .. 3MB)
//   K3 gather : async global->LDS stage of blended row (ASYNCcnt +
//               s_wait_asynccnt), per-pixel LDS gather + fma, non-temporal
//               coalesced b128 stores (output is 256MB write-once > 192MB L2).
// Roofline: 256 MB stores @ 23.3 TB/s => ~11 us floor; everything else L2-hot.
// ---------------------------------------------------------------------------

#define Hr 64
#define Wr 2048
#define Hb 512
#define Wb 512
#define NPIX (Hb * Wb)          // 262144
#define NIMG 256                // B*C
#define THREADS 256
#define TILES 8
#define TILE (NPIX / TILES)     // 32768 pixels per block
#define GROUPS_PER_THREAD (TILE / (4 * THREADS))  // 32 float4 groups

#define TWO_PI_D 6.283185307179586

// ext-vector types (HIP's float4/int4 are structs; builtins need ext-vectors)
typedef int   v4i __attribute__((ext_vector_type(4)));
typedef float v4f __attribute__((ext_vector_type(4)));
typedef __attribute__((address_space(1))) v4i* g_b128_ptr;   // global src
typedef __attribute__((address_space(3))) v4i* l_b128_ptr;   // LDS dst

// ---------------------------------------------------------------------------
// K1: px coordinate table (fp64 math mirroring numpy, cast to fp32).
// ---------------------------------------------------------------------------
__global__ void rv2bev_coords_kernel(float* __restrict__ px_tab) {
    int n = blockIdx.x * blockDim.x + threadIdx.x;   // 0 .. 262143
    int yy = n >> 9;          // row    (meshgrid 'ij')
    int xx = n & 511;         // column
    double y = (((double)yy - 256.0 + 0.5) * 50.0) / 255.5;
    double x = (((double)xx - 256.0 + 0.5) * 50.0) / 255.5;
    double phi = fmod(atan2(y, x) + TWO_PI_D, TWO_PI_D);
    double col = (double)(Wr - 1) - phi / TWO_PI_D * (double)(Wr - 1);
    double gx  = col / (double)Wr * 2.0 - 1.0;
    double px  = (gx + 1.0) / 2.0 * (double)(Wr - 1);
    px_tab[n] = (float)px;
}

// ---------------------------------------------------------------------------
// K2: blended[img][w] = 0.5*rv[img][31][w] + 0.5*rv[img][32][w]  (float4)
// ---------------------------------------------------------------------------
__global__ void rv2bev_blend_kernel(const float* __restrict__ rv,
                                    float* __restrict__ blended) {
    int t = blockIdx.x * blockDim.x + threadIdx.x;   // float4 index, 0..131071
    int img = t >> 9;                                // 512 float4 per row
    int w4  = t & 511;
    const v4f* r0 = (const v4f*)(rv + ((size_t)img * Hr + (Hr / 2 - 1)) * Wr);
    const v4f* r1 = r0 + Wr / 4;
    v4f v = 0.5f * r0[w4] + 0.5f * r1[w4];
    ((v4f*)blended)[t] = v;
}

// ---------------------------------------------------------------------------
// K3: stage blended row -> LDS (async b128), gather + x-lerp, NT stores.
// ---------------------------------------------------------------------------
__device__ __forceinline__ float lds_lerp(const float* __restrict__ row, float px) {
    float f  = floorf(px);
    int   x0 = (int)f;
    x0 = x0 < 0 ? 0 : (x0 > Wr - 2 ? Wr - 2 : x0);
    float wx = px - f;
    float s0 = row[x0];
    float s1 = row[x0 + 1];
    return fmaf(wx, s1 - s0, s0);   // s0*(1-wx) + s1*wx
}

__global__ void __launch_bounds__(THREADS)
rv2bev_gather_kernel(const float* __restrict__ blended,
                     const float* __restrict__ px_tab,
                     float* __restrict__ out) {
    __shared__ float sRow[Wr];    // 8 KB: the wy-blended row for this image

    const int img = blockIdx.y;                      // b*C + c, 0..255
    const float* g0 = blended + (size_t)img * Wr;

    // ---- Stage blended row into LDS (512 x b128 chunks) -------------------
#if defined(__HIP_DEVICE_COMPILE__) && __has_builtin(__builtin_amdgcn_global_load_async_to_lds_b128)
    for (int i = threadIdx.x; i < Wr / 4; i += THREADS) {   // 16B chunks
        __builtin_amdgcn_global_load_async_to_lds_b128(
            (g_b128_ptr)(g0 + i * 4), (l_b128_ptr)(&sRow[i * 4]), 0, 0);
    }
#if __has_builtin(__builtin_amdgcn_s_wait_asynccnt)
    __builtin_amdgcn_s_wait_asynccnt(0);
#else
    asm volatile("s_wait_asynccnt 0" ::: "memory");
#endif
#else
    for (int i = threadIdx.x; i < Wr; i += THREADS)
        sRow[i] = g0[i];
#endif
    __syncthreads();

    // ---- Gather + x-lerp, non-temporal float4-coalesced stores ------------
    const v4f* __restrict__ px4  = (const v4f*)px_tab;
    v4f* __restrict__       out4 = (v4f*)(out + (size_t)img * NPIX);
    const int qbase = (blockIdx.x * TILE) / 4;

#pragma unroll 4
    for (int k = 0; k < GROUPS_PER_THREAD; ++k) {
        int q = qbase + k * THREADS + threadIdx.x;   // float4 index
        v4f p = px4[q];
        v4f r;
        r.x = lds_lerp(sRow, p.x);
        r.y = lds_lerp(sRow, p.y);
        r.z = lds_lerp(sRow, p.z);
        r.w = lds_lerp(sRow, p.w);
        __builtin_nontemporal_store(r, &out4[q]);    // TH=NT: bypass L2 churn
    }
}

// ---------------------------------------------------------------------------
// Launch
// ---------------------------------------------------------------------------
extern "C" void kernel_launch(void* const* d_in, const int* in_sizes, int n_in,
                              void* d_out, int out_size, void* d_ws, size_t ws_size,
                              hipStream_t stream) {
    const float* rv_feat = (const float*)d_in[0];   // (4,64,64,2048) fp32
    // d_in[1] (ref_bev) is unused by the reference output.
    float* out     = (float*)d_out;                 // (4,64,512,512) fp32
    float* px_tab  = (float*)d_ws;                  // [0, 1MB): 262144 floats
    float* blended = px_tab + NPIX;                 // [1MB, 3MB): 256*2048 floats

    rv2bev_coords_kernel<<<NPIX / THREADS, THREADS, 0, stream>>>(px_tab);

    rv2bev_blend_kernel<<<(NIMG * Wr / 4) / THREADS, THREADS, 0, stream>>>(
        rv_feat, blended);

    dim3 grid(TILES, NIMG);
    rv2bev_gather_kernel<<<grid, THREADS, 0, stream>>>(blended, px_tab, out);

    (void)in_sizes; (void)n_in; (void)out_size; (void)ws_size;
}